// LevelLatencyEncoder_25323127177876
// MI455X (gfx1250) — compile-verified
//
#include <hip/hip_runtime.h>
#include <math.h>

#define NN     100000
#define NE     1600000
#define EMB3   192
#define F_LAT  64
#define F_OUT  128
#define NEG_SLOPE 0.2f

typedef __attribute__((ext_vector_type(2))) float v2f;
typedef __attribute__((ext_vector_type(8))) float v8f;

// D = A(16x4 f32) * B(4x16 f32) + C(16x16 f32)   -- full fp32 matrix core
__device__ __forceinline__ v8f wmma4(v2f a, v2f b, v8f c) {
  return __builtin_amdgcn_wmma_f32_16x16x4_f32(false, a, false, b, (short)0, c,
                                               false, false);
}

// order-preserving f32 <-> u32 encoding so segment-max can use atomicMax(u32)
__device__ __forceinline__ unsigned enc_f32(float f) {
  unsigned u = __float_as_uint(f);
  return (u & 0x80000000u) ? ~u : (u | 0x80000000u);
}
__device__ __forceinline__ float dec_f32(unsigned u) {
  return (u & 0x80000000u) ? __uint_as_float(u & 0x7FFFFFFFu)
                           : __uint_as_float(~u);
}

// ---------------------------------------------------------------------------
// K0: zero rst / denom / emax
__global__ void init_kernel(float* __restrict__ rst, float* __restrict__ denom,
                            unsigned* __restrict__ emax_u) {
  int i = blockIdx.x * blockDim.x + threadIdx.x;
  if (i < NN * F_OUT) rst[i] = 0.0f;
  if (i < NN) { denom[i] = 0.0f; emax_u[i] = 0u; }
}

// ---------------------------------------------------------------------------
// K1: per 16-node tile: x = relu(embed@W_lin + b); feat = x@W_gat;
//     el = feat@attn_l; er = feat@attn_r.   One wave32 per tile, WMMA f32.
__global__ __launch_bounds__(256) void node_feat_kernel(
    const float* __restrict__ embed,
    const float* __restrict__ W_lin, const float* __restrict__ b_lin,
    const float* __restrict__ W_gat,
    const float* __restrict__ attn_l, const float* __restrict__ attn_r,
    float* __restrict__ feat, float* __restrict__ el, float* __restrict__ er,
    int n_tiles) {
  extern __shared__ float smem[];
  float* sW1 = smem;                    // 192*64
  float* sW2 = sW1 + EMB3 * F_LAT;      // 64*128
  float* sB1 = sW2 + F_LAT * F_OUT;     // 64
  float* sAl = sB1 + F_LAT;             // 128
  float* sAr = sAl + F_OUT;             // 128
  float* sX  = sAr + F_OUT;             // 8 waves * 16*64

  const int tid = threadIdx.x;
  for (int i = tid; i < EMB3 * F_LAT; i += 256) sW1[i] = W_lin[i];
  for (int i = tid; i < F_LAT * F_OUT; i += 256) sW2[i] = W_gat[i];
  if (tid < F_LAT) sB1[tid] = b_lin[tid];
  if (tid < F_OUT) { sAl[tid] = attn_l[tid]; sAr[tid] = attn_r[tid]; }
  __syncthreads();

  const int wave = tid >> 5;
  const int lane = tid & 31;
  const int tile = blockIdx.x * 8 + wave;
  if (tile >= n_tiles) return;          // wave-uniform exit; EXEC stays all-1s

  const int m  = lane & 15;             // A: row M;  B/D: col N
  const int hi = lane >> 4;             // selects K pair {0,1} vs {2,3}

  // ---- GEMM1: 16x192 @ 192x64, K-loop of 48 WMMA steps x 4 col tiles ----
  const float* arow = embed + (size_t)(tile * 16 + m) * EMB3;
  v8f acc[4] = {};
  for (int k0 = 0; k0 < EMB3; k0 += 4) {
    const int ka = k0 + 2 * hi;
    v2f a; a.x = arow[ka]; a.y = arow[ka + 1];
    const float* b0 = sW1 + ka * F_LAT + m;
    const float* b1 = b0 + F_LAT;
#pragma unroll
    for (int nt = 0; nt < 4; ++nt) {
      v2f b; b.x = b0[nt * 16]; b.y = b1[nt * 16];
      acc[nt] = wmma4(a, b, acc[nt]);
    }
  }

  // bias + relu -> per-wave LDS x tile (D-fragment -> A-fragment bounce)
  float* xw = sX + wave * 16 * F_LAT;
#pragma unroll
  for (int r = 0; r < 8; ++r) {
    const int row = r + 8 * hi;
#pragma unroll
    for (int nt = 0; nt < 4; ++nt) {
      float v = acc[nt][r] + sB1[nt * 16 + m];
      xw[row * F_LAT + nt * 16 + m] = fmaxf(v, 0.0f);
    }
  }

  // ---- GEMM2: 16x64 @ 64x128, 16 WMMA steps x 8 col tiles ----
  v8f fa[8] = {};
  for (int k0 = 0; k0 < F_LAT; k0 += 4) {
    const int ka = k0 + 2 * hi;
    v2f a; a.x = xw[m * F_LAT + ka]; a.y = xw[m * F_LAT + ka + 1];
    const float* b0 = sW2 + ka * F_OUT + m;
    const float* b1 = b0 + F_OUT;
#pragma unroll
    for (int nt = 0; nt < 8; ++nt) {
      v2f b; b.x = b0[nt * 16]; b.y = b1[nt * 16];
      fa[nt] = wmma4(a, b, fa[nt]);
    }
  }

  // write feat; accumulate el/er partial dot-products in registers
  float pl[8], pr[8];
  const size_t base = (size_t)(tile * 16) * F_OUT;
#pragma unroll
  for (int r = 0; r < 8; ++r) {
    const int row = r + 8 * hi;
    float sl = 0.0f, sr = 0.0f;
#pragma unroll
    for (int nt = 0; nt < 8; ++nt) {
      const float v = fa[nt][r];
      feat[base + row * F_OUT + nt * 16 + m] = v;
      sl += v * sAl[nt * 16 + m];
      sr += v * sAr[nt * 16 + m];
    }
    pl[r] = sl; pr[r] = sr;
  }
  // reduce across the 16 lanes of each half (xor<=8 never crosses lane bit 4)
#pragma unroll
  for (int off = 8; off >= 1; off >>= 1) {
#pragma unroll
    for (int r = 0; r < 8; ++r) {
      pl[r] += __shfl_xor(pl[r], off, 32);
      pr[r] += __shfl_xor(pr[r], off, 32);
    }
  }
  if (m == 0) {
#pragma unroll
    for (int r = 0; r < 8; ++r) {
      el[tile * 16 + r + 8 * hi] = pl[r];
      er[tile * 16 + r + 8 * hi] = pr[r];
    }
  }
}

// ---------------------------------------------------------------------------
// K2: per-edge leaky-relu logit + encoded segment max into emax_u[dst]
__global__ void edge_logit_kernel(const int* __restrict__ src,
                                  const int* __restrict__ dst,
                                  const float* __restrict__ el,
                                  const float* __restrict__ er,
                                  float* __restrict__ e_edge,
                                  unsigned* __restrict__ emax_u) {
  int i = blockIdx.x * blockDim.x + threadIdx.x;
  if (i >= NE) return;
  float e = el[src[i]] + er[dst[i]];
  e = (e > 0.0f) ? e : NEG_SLOPE * e;
  e_edge[i] = e;
  atomicMax(emax_u + dst[i], enc_f32(e));
}

// K3: ee = exp(e - emax[dst]); denom[dst] += ee
__global__ void edge_exp_kernel(const int* __restrict__ dst,
                                float* __restrict__ e_edge,
                                const unsigned* __restrict__ emax_u,
                                float* __restrict__ denom) {
  int i = blockIdx.x * blockDim.x + threadIdx.x;
  if (i >= NE) return;
  const int d = dst[i];
  float emax = dec_f32(emax_u[d]);
  if ((__float_as_uint(emax) & 0x7F800000u) == 0x7F800000u) emax = 0.0f;
  float ee = expf(e_edge[i] - emax);
  e_edge[i] = ee;
  atomicAdd(denom + d, ee);
}

// K4: one wave per edge; float4 gather of feat[src] (L2-resident),
//     4x global_atomic_add_f32 per lane into rst[dst] (L2-resident)
__global__ __launch_bounds__(256) void edge_scatter_kernel(
    const int* __restrict__ src, const int* __restrict__ dst,
    const float* __restrict__ e_edge, const float* __restrict__ denom,
    const float* __restrict__ feat, float* __restrict__ rst) {
  int gid = blockIdx.x * blockDim.x + threadIdx.x;
  int e = gid >> 5;
  int lane = gid & 31;
  if (e >= NE) return;
  const int s = src[e], d = dst[e];
  float dn = denom[d];
  dn = (dn > 0.0f) ? dn : 1.0f;
  const float alpha = e_edge[e] / dn;
  const float4 v = ((const float4*)(feat + (size_t)s * F_OUT))[lane];
  float* rr = rst + (size_t)d * F_OUT + lane * 4;
  atomicAdd(rr + 0, v.x * alpha);
  atomicAdd(rr + 1, v.y * alpha);
  atomicAdd(rr + 2, v.z * alpha);
  atomicAdd(rr + 3, v.w * alpha);
}

// K5: y = rst + b_gat; out = [mu (N x 64) | tanh(logvar) (N x 64)] flat
__global__ void finalize_kernel(const float* __restrict__ rst,
                                const float* __restrict__ b_gat,
                                float* __restrict__ out) {
  int i = blockIdx.x * blockDim.x + threadIdx.x;
  if (i >= NN * F_OUT) return;
  const int n = i >> 7, c = i & 127;
  const float y = rst[i] + b_gat[c];
  if (c < 64) out[(size_t)n * 64 + c] = y;
  else        out[(size_t)NN * 64 + (size_t)n * 64 + (c - 64)] = tanhf(y);
}

// ---------------------------------------------------------------------------
extern "C" void kernel_launch(void* const* d_in, const int* in_sizes, int n_in,
                              void* d_out, int out_size, void* d_ws,
                              size_t ws_size, hipStream_t stream) {
  const float* embed  = (const float*)d_in[0];
  const int*   src    = (const int*)d_in[1];   // jax x64 disabled -> int32
  const int*   dst    = (const int*)d_in[2];
  const float* W_lin  = (const float*)d_in[3];
  const float* b_lin  = (const float*)d_in[4];
  const float* W_gat  = (const float*)d_in[5];
  const float* attn_l = (const float*)d_in[6];
  const float* attn_r = (const float*)d_in[7];
  const float* b_gat  = (const float*)d_in[8];
  float* out = (float*)d_out;

  // workspace layout (floats): feat | el | er | denom | emax_u | e_edge | rst
  float* ws     = (float*)d_ws;
  float* feat   = ws;                              // NN*128
  float* el     = feat + (size_t)NN * F_OUT;       // NN
  float* er     = el + NN;                         // NN
  float* denom  = er + NN;                         // NN
  unsigned* emax_u = (unsigned*)(denom + NN);      // NN
  float* e_edge = (float*)(emax_u + NN);           // NE
  float* rst    = e_edge + NE;                     // NN*128  (~110 MB total)

  init_kernel<<<(NN * F_OUT + 255) / 256, 256, 0, stream>>>(rst, denom, emax_u);

  const int n_tiles = NN / 16;                     // 6250 (exact)
  const int nf_blocks = (n_tiles + 7) / 8;
  const size_t smem_bytes =
      (size_t)(EMB3 * F_LAT + F_LAT * F_OUT + F_LAT + 2 * F_OUT +
               8 * 16 * F_LAT) * sizeof(float);    // 115,968 B (< 320 KB WGP LDS)
  (void)hipFuncSetAttribute((const void*)node_feat_kernel,
                            hipFuncAttributeMaxDynamicSharedMemorySize,
                            (int)smem_bytes);
  node_feat_kernel<<<nf_blocks, 256, smem_bytes, stream>>>(
      embed, W_lin, b_lin, W_gat, attn_l, attn_r, feat, el, er, n_tiles);

  edge_logit_kernel<<<(NE + 255) / 256, 256, 0, stream>>>(src, dst, el, er,
                                                          e_edge, emax_u);
  edge_exp_kernel<<<(NE + 255) / 256, 256, 0, stream>>>(dst, e_edge, emax_u,
                                                        denom);
  edge_scatter_kernel<<<(NE * 32 + 255) / 256, 256, 0, stream>>>(
      src, dst, e_edge, denom, feat, rst);
  finalize_kernel<<<(NN * F_OUT + 255) / 256, 256, 0, stream>>>(rst, b_gat,
                                                                out);
}